// RotationHopfieldNetwork2D_81905026335369
// MI455X (gfx1250) — compile-verified
//
#include <hip/hip_runtime.h>

typedef __attribute__((ext_vector_type(2))) float v2f;
typedef __attribute__((ext_vector_type(8))) float v8f;

#define LR_C 0.01f
#define LOG2E 1.4426950408889634f
#define WAVES 8
#define ROWS_PER_WAVE 16
#define NPAT 128
#define NTILE 8  // 128 patterns / 16 cols per WMMA

__global__ __launch_bounds__(WAVES * 32)
void hopfield2d_kernel(const float* __restrict__ x_in,
                       const float* __restrict__ patterns,
                       const unsigned char* __restrict__ mask,
                       const int* __restrict__ max_iter_p,
                       float* __restrict__ x_out,
                       int nrows)
{
    // phi(patterns): 128 x (f1,f2,f3,pad)
    __shared__ float phiP[NPAT][4];
    // per-wave softmax partials: [row][col][se,w1,w2,w3], row stride 17 -> conflict-free b128
    __shared__ float part[WAVES][ROWS_PER_WAVE * 17][4];

    const int tid  = threadIdx.x;
    const int lane = tid & 31;
    const int wave = tid >> 5;
    const int half = lane >> 4;   // which 16-lane half
    const int l16  = lane & 15;

    if (tid < NPAT) {
        float p0 = patterns[2 * tid], p1 = patterns[2 * tid + 1];
        phiP[tid][0] = p0 * p0 + p1 * p1;
        phiP[tid][1] = (p0 * p0) * (p1 * p1);
        phiP[tid][2] = p0 * p1 * (p0 * p0 - p1 * p1);
        phiP[tid][3] = 0.0f;
    }
    __syncthreads();

    // B tiles (4x16 f32): PRE-SCALED by log2(e) so WMMA emits log2-domain logits.
    // pf* arrays (for the softmax-weighted sums) stay unscaled.
    v2f Bt[NTILE];
    float pf1[NTILE], pf2[NTILE], pf3[NTILE];
#pragma unroll
    for (int t = 0; t < NTILE; ++t) {
        int n = t * 16 + l16;
        pf1[t] = phiP[n][0];
        pf2[t] = phiP[n][1];
        pf3[t] = phiP[n][2];
        Bt[t].x = (half ? pf3[t] : pf1[t]) * LOG2E;  // K2 : K0
        Bt[t].y = half ? 0.0f : (pf2[t] * LOG2E);    // K3 : K1
    }

    const int row0  = (blockIdx.x * WAVES + wave) * ROWS_PER_WAVE;
    const int myrow = row0 + l16;            // duplicated across halves
    const bool valid = (myrow < nrows);
    const int rc = valid ? myrow : (nrows - 1);

    float x0 = x_in[2 * rc];
    float x1 = x_in[2 * rc + 1];
    const bool m0 = mask[2 * rc] != 0;
    const bool m1 = mask[2 * rc + 1] != 0;
    const int iters = *max_iter_p;

    for (int k = 0; k < iters; ++k) {
        float x0s = x0 * x0, x1s = x1 * x1;
        float f1 = x0s + x1s;
        float f2 = x0s * x1s;
        float f3 = x0 * x1 * (x0s - x1s);

        // A 16x4 f32: lanes 0-15 -> (K0,K1)=(f1,f2); lanes 16-31 -> (K2,K3)=(f3,0)
        v2f A;
        A.x = half ? f3 : f1;
        A.y = half ? 0.0f : f2;

        // ---- logits (log2 domain), 8 tiles of 16x16 ----
        v8f Ct[NTILE];
#pragma unroll
        for (int t = 0; t < NTILE; ++t) {
            Ct[t] = __builtin_amdgcn_wmma_f32_16x16x4_f32(
                false, A, false, Bt[t],
                (short)0, (v8f){0.f, 0.f, 0.f, 0.f, 0.f, 0.f, 0.f, 0.f},
                false, false);
        }

        // ---- row max (log2 domain): per-reg tile max, then 16-lane xor-reduce ----
        float rm[8];
#pragma unroll
        for (int r = 0; r < 8; ++r) {
            float v = Ct[0][r];
#pragma unroll
            for (int t = 1; t < NTILE; ++t) v = fmaxf(v, Ct[t][r]);
            rm[r] = v;
        }
#pragma unroll
        for (int r = 0; r < 8; ++r) {
            rm[r] = fmaxf(rm[r], __shfl_xor(rm[r], 1, 32));
            rm[r] = fmaxf(rm[r], __shfl_xor(rm[r], 2, 32));
            rm[r] = fmaxf(rm[r], __shfl_xor(rm[r], 4, 32));
            rm[r] = fmaxf(rm[r], __shfl_xor(rm[r], 8, 32));
        }

        // ---- exp2 + weighted partial sums ----
        float se[8], w1[8], w2[8], w3[8];
#pragma unroll
        for (int r = 0; r < 8; ++r) { se[r] = 0.f; w1[r] = 0.f; w2[r] = 0.f; w3[r] = 0.f; }
#pragma unroll
        for (int t = 0; t < NTILE; ++t) {
#pragma unroll
            for (int r = 0; r < 8; ++r) {
                float e = __builtin_amdgcn_exp2f(Ct[t][r] - rm[r]);
                se[r] += e;
                w1[r] += e * pf1[t];
                w2[r] += e * pf2[t];
                w3[r] += e * pf3[t];
            }
        }

        // ---- 16-lane row reduction + layout transpose via padded LDS ----
#pragma unroll
        for (int r = 0; r < 8; ++r) {
            int m = r + 8 * half;             // C-layout row of this reg
            float4 v = make_float4(se[r], w1[r], w2[r], w3[r]);
            *(float4*)&part[wave][m * 17 + l16][0] = v;
        }
        __syncthreads();   // store -> load ordering (per-wave region; single barrier suffices:
                           // DS is in-order per wave, so next-iter stores cannot pass these loads)
        float SE = 0.f, W1 = 0.f, W2 = 0.f, W3 = 0.f;
#pragma unroll
        for (int c = 0; c < 16; ++c) {
            float4 v = *(const float4*)&part[wave][l16 * 17 + c][0];
            SE += v.x; W1 += v.y; W2 += v.z; W3 += v.w;
        }

        // ---- gradient + update (lane owns row l16; both halves duplicate) ----
        float inv = __builtin_amdgcn_rcpf(SE);
        float g1 = f1 - W1 * inv;
        float g2 = f2 - W2 * inv;
        float g3 = f3 - W3 * inv;
        float gx0 = 2.f * x0 * g1 + 2.f * x0 * x1s * g2 + (3.f * x0s * x1 - x1s * x1) * g3;
        float gx1 = 2.f * x1 * g1 + 2.f * x0s * x1 * g2 + (x0s * x0 - 3.f * x0 * x1s) * g3;
        float nx0 = x0 - LR_C * gx0;
        float nx1 = x1 - LR_C * gx1;
        x0 = m0 ? nx0 : x0;
        x1 = m1 ? nx1 : x1;
    }

    if (valid && half == 0) {
        x_out[2 * myrow]     = x0;
        x_out[2 * myrow + 1] = x1;
    }
}

extern "C" void kernel_launch(void* const* d_in, const int* in_sizes, int n_in,
                              void* d_out, int out_size, void* d_ws, size_t ws_size,
                              hipStream_t stream) {
    const float* x            = (const float*)d_in[0];
    const float* patterns     = (const float*)d_in[1];
    const unsigned char* mask = (const unsigned char*)d_in[2];
    const int* max_iter       = (const int*)d_in[3];
    float* out                = (float*)d_out;

    int nrows = in_sizes[0] / 2;
    int rows_per_block = WAVES * ROWS_PER_WAVE;
    int blocks = (nrows + rows_per_block - 1) / rows_per_block;

    hipLaunchKernelGGL(hopfield2d_kernel, dim3(blocks), dim3(WAVES * 32), 0, stream,
                       x, patterns, mask, max_iter, out, nrows);
}